// GCNScoreModel_901943132623
// MI455X (gfx1250) — compile-verified
//
#include <hip/hip_runtime.h>

typedef __attribute__((ext_vector_type(2))) float v2f;
typedef __attribute__((ext_vector_type(8))) float v8f;

#define TPB 256

// ---------------------------------------------------------------- degree prep
__global__ void k_deg_init(float* deg, int n) {
    int t = blockIdx.x * blockDim.x + threadIdx.x;
    if (t < n) deg[t] = 1.0f;   // self loop
}

__global__ void k_deg_accum(const int* __restrict__ dst, float* deg, int e) {
    int t = blockIdx.x * blockDim.x + threadIdx.x;
    if (t < e) unsafeAtomicAdd(&deg[dst[t]], 1.0f);
}

__global__ void k_deg_rsqrt(float* deg, int n) {
    int t = blockIdx.x * blockDim.x + threadIdx.x;
    if (t < n) deg[t] = rsqrtf(deg[t]);   // deg >= 1 always (self loop)
}

// ---------------------------------------------------------------- WMMA GEMM
// H[N x 32] = act(X[N x 32]) @ W[32 x 32]; act = relu if RELU.
// Fused epilogue: G[i][f] = H[i][f]*dinv[i]^2 + b[f]  (self-loop msg + bias init).
// One wave per 16-row tile; two 16-col output tiles; f32 WMMA 16x16x4, 8 k-steps.
template <bool RELU>
__global__ void k_gemm32(const float* __restrict__ X, const float* __restrict__ W,
                         const float* __restrict__ dinv, const float* __restrict__ b,
                         float* __restrict__ H, float* __restrict__ G, int nrows) {
    int wave = (blockIdx.x * blockDim.x + threadIdx.x) >> 5;   // wave-uniform
    int lane = threadIdx.x & 31;
    int row0 = wave * 16;
    if (row0 >= nrows) return;                                  // whole-wave exit

    int m  = lane & 15;      // A row within tile / B,C column
    int hi = lane >> 4;      // half-wave selector

    const float* xrow = X + (size_t)(row0 + m) * 32;
    v8f c0 = {};
    v8f c1 = {};

#pragma unroll
    for (int kb = 0; kb < 8; ++kb) {
        int k = kb * 4 + hi * 2;                 // lanes 0-15: K,K+1 ; 16-31: K+2,K+3
        float ax = xrow[k];
        float ay = xrow[k + 1];
        if (RELU) {
            ax = ax > 0.0f ? ax : 0.0f;
            ay = ay > 0.0f ? ay : 0.0f;
        }
        v2f a = { ax, ay };
        v2f b0 = { W[k * 32 + m],       W[(k + 1) * 32 + m]      };
        v2f b1 = { W[k * 32 + 16 + m],  W[(k + 1) * 32 + 16 + m] };
        c0 = __builtin_amdgcn_wmma_f32_16x16x4_f32(false, a, false, b0, (short)0, c0, false, false);
        c1 = __builtin_amdgcn_wmma_f32_16x16x4_f32(false, a, false, b1, (short)0, c1, false, false);
    }

    float bias0 = b[m];
    float bias1 = b[16 + m];
#pragma unroll
    for (int v = 0; v < 8; ++v) {
        int r = row0 + v + 8 * hi;               // C layout: VGPR v -> M = v (+8 for hi lanes)
        float di = dinv[r];
        float d2 = di * di;
        float h0 = c0[v];
        float h1 = c1[v];
        H[(size_t)r * 32 + m]      = h0;
        H[(size_t)r * 32 + 16 + m] = h1;
        G[(size_t)r * 32 + m]      = h0 * d2 + bias0;
        G[(size_t)r * 32 + 16 + m] = h1 * d2 + bias1;
    }
}

// ---------------------------------------------------------------- edge scatter
// one wave per edge, lane = feature: G[dst] += H[src]*dinv[src]*dinv[dst]
__global__ void k_scatter32(const float* __restrict__ H, const int* __restrict__ src,
                            const int* __restrict__ dst, const float* __restrict__ dinv,
                            float* __restrict__ G, int e) {
    int t = blockIdx.x * blockDim.x + threadIdx.x;
    int ed = t >> 5, f = t & 31;
    if (ed >= e) return;
    int s = src[ed], d = dst[ed];
    float nrm = dinv[s] * dinv[d];
    unsafeAtomicAdd(&G[(size_t)d * 32 + f], H[(size_t)s * 32 + f] * nrm);
}

// ---------------------------------------------------------------- final layer
// S[i] = sum_f relu(G[i][f]) * W2[f]  (wave per node, shuffle reduce)
// Fused: Out[i] = S[i]*dinv[i]^2 + b2[0]  (self-loop + bias init)
__global__ void k_dot32(const float* __restrict__ G, const float* __restrict__ W2,
                        const float* __restrict__ dinv, const float* __restrict__ b2,
                        float* __restrict__ S, float* __restrict__ Out, int n) {
    int t = blockIdx.x * blockDim.x + threadIdx.x;
    int i = t >> 5, f = t & 31;
    if (i >= n) return;
    float g = G[(size_t)i * 32 + f];
    float v = (g > 0.0f ? g : 0.0f) * W2[f];
#pragma unroll
    for (int off = 16; off > 0; off >>= 1) v += __shfl_xor(v, off, 32);
    if (f == 0) {
        S[i] = v;
        float di = dinv[i];
        Out[i] = v * di * di + b2[0];
    }
}

__global__ void k_final_scatter(const int* __restrict__ src, const int* __restrict__ dst,
                                const float* __restrict__ dinv, const float* __restrict__ S,
                                float* __restrict__ Out, int e) {
    int t = blockIdx.x * blockDim.x + threadIdx.x;
    if (t >= e) return;
    int s = src[t], d = dst[t];
    unsafeAtomicAdd(&Out[d], S[s] * dinv[s] * dinv[d]);
}

// ---------------------------------------------------------------- launch
extern "C" void kernel_launch(void* const* d_in, const int* in_sizes, int n_in,
                              void* d_out, int out_size, void* d_ws, size_t ws_size,
                              hipStream_t stream) {
    const float* x  = (const float*)d_in[0];
    const int*   ei = (const int*)  d_in[1];
    const float* W0 = (const float*)d_in[2];
    const float* b0 = (const float*)d_in[3];
    const float* W1 = (const float*)d_in[4];
    const float* b1 = (const float*)d_in[5];
    const float* W2 = (const float*)d_in[6];
    const float* b2 = (const float*)d_in[7];
    float* out = (float*)d_out;

    int N = in_sizes[0] / 32;
    int E = in_sizes[1] / 2;
    const int* src = ei;
    const int* dst = ei + E;

    float* ws   = (float*)d_ws;
    float* dinv = ws;
    float* bufH = ws + (size_t)((N + 15) & ~15);
    float* bufG = bufH + (size_t)N * 32;
    float* S    = bufH;   // reused after bufH's last read

    dim3 blk(TPB);
    dim3 gN((N + TPB - 1) / TPB);
    dim3 gE((E + TPB - 1) / TPB);
    int gemmThreads = ((N + 15) / 16) * 32;
    dim3 gGemm((gemmThreads + TPB - 1) / TPB);
    dim3 gNF(((size_t)N * 32 + TPB - 1) / TPB);
    dim3 gEF(((size_t)E * 32 + TPB - 1) / TPB);

    // normalization
    k_deg_init  <<<gN, blk, 0, stream>>>(dinv, N);
    k_deg_accum <<<gE, blk, 0, stream>>>(dst, dinv, E);
    k_deg_rsqrt <<<gN, blk, 0, stream>>>(dinv, N);

    // layer 0: H = x@W0 ; G = H*dinv^2 + b0 ; G += edge messages
    k_gemm32<false><<<gGemm, blk, 0, stream>>>(x, W0, dinv, b0, bufH, bufG, N);
    k_scatter32    <<<gEF,   blk, 0, stream>>>(bufH, src, dst, dinv, bufG, E);

    // layer 1: H = relu(G)@W1 ; G = H*dinv^2 + b1 ; G += edge messages
    k_gemm32<true> <<<gGemm, blk, 0, stream>>>(bufG, W1, dinv, b1, bufH, bufG, N);
    k_scatter32    <<<gEF,   blk, 0, stream>>>(bufH, src, dst, dinv, bufG, E);

    // layer 2: S = relu(G)@W2 ; Out = S*dinv^2 + b2 ; Out += edge messages
    k_dot32        <<<gNF, blk, 0, stream>>>(bufG, W2, dinv, b2, S, out, N);
    k_final_scatter<<<gE,  blk, 0, stream>>>(src, dst, dinv, S, out, E);
}